// GTS_Model_38268158607754
// MI455X (gfx1250) — compile-verified
//
#include <hip/hip_runtime.h>
#include <hip/hip_bf16.h>
#include <math.h>

// ---- model constants (must match reference) ----
#define N_NODES  512
#define EMB      32
#define N_EDGES  26000
#define TOTAL_T  4096
#define WINDOW   64
#define SLIDE    64
#define N_WIN    63          // (4096-64)/64
#define C1C      32
#define C2C      16
#define K1C      8
#define K2C      8
#define S1C      4
#define S2C      4
#define L1C      15          // (64-8)/4+1
#define L2C      2           // (15-8)/4+1
#define EPSF     1e-5f
#define KSTACK   320         // stacked diffusion states: [xh, t1o, t1i, t2o, t2i] * 64
#define H2DIM    8192
#define H8DIM    2048
#define REGROWS  1484        // 2*742
#define CLS2ROWS 8

typedef float v2f __attribute__((ext_vector_type(2)));
typedef float v8f __attribute__((ext_vector_type(8)));

// ---------------- utility ----------------
__global__ void zero_kernel(float* p, int n) {
  int i = blockIdx.x * blockDim.x + threadIdx.x;
  if (i < n) p[i] = 0.f;
}

// ---------------- graph degree / inverse ----------------
__global__ void deg_kernel(const int* __restrict__ ei, float* deg_out, float* deg_in) {
  int e = blockIdx.x * blockDim.x + threadIdx.x;
  if (e < N_EDGES) {
    atomicAdd(&deg_out[ei[e]], 1.f);
    atomicAdd(&deg_in[ei[N_EDGES + e]], 1.f);
  }
}
__global__ void inv_kernel(float* d, int n) {
  int i = blockIdx.x * blockDim.x + threadIdx.x;
  if (i < n) d[i] = (d[i] > 0.f) ? 1.f / d[i] : 0.f;
}

// ---------------- combined diffusion weights ----------------
// W is (2,3,64,32) row-major. Stacked K rows: block 0 -> W[0,0]+W[1,0],
// 1 -> W[0,1], 2 -> W[1,1], 3 -> W[0,2], 4 -> W[1,2].
__device__ __forceinline__ float combw(const float* W, int kb, int f, int c) {
  int idx = f * 32 + c;
  switch (kb) {
    case 0:  return W[idx] + W[3 * 2048 + idx];
    case 1:  return W[1 * 2048 + idx];
    case 2:  return W[4 * 2048 + idx];
    case 3:  return W[2 * 2048 + idx];
    default: return W[5 * 2048 + idx];
  }
}
__global__ void wcomb_kernel(const float* __restrict__ wz, const float* __restrict__ wr,
                             const float* __restrict__ wh, float* Wzr, float* Whc) {
  int i = blockIdx.x * blockDim.x + threadIdx.x;   // 320*64 + 320*32 total
  if (i < KSTACK * 64) {
    int k = i / 64, n = i % 64;
    const float* W = (n < 32) ? wz : wr;
    int c = n & 31;
    Wzr[k * 64 + n] = combw(W, k / 64, k % 64, c);
  } else if (i < KSTACK * 96) {
    int j = i - KSTACK * 64;
    int k = j / 32, c = j % 32;
    Whc[k * 32 + c] = combw(wh, k / 64, k % 64, c);
  }
}

// ---------------- conv1 + relu ----------------
__global__ void conv1_kernel(const float* __restrict__ inputs, const float* __restrict__ w,
                             const float* __restrict__ b, float* __restrict__ A, int start) {
  __shared__ float win[WINDOW];
  int n = blockIdx.x;
  int tid = threadIdx.x;                 // blockDim = 480 = C1C*L1C
  if (tid < WINDOW) win[tid] = inputs[n * TOTAL_T + start + tid];
  __syncthreads();
  int c = tid / L1C, p = tid % L1C;
  float acc = b[c];
#pragma unroll
  for (int k = 0; k < K1C; k++) acc += w[c * K1C + k] * win[p * S1C + k];
  A[(n * C1C + c) * L1C + p] = fmaxf(acc, 0.f);
}

// ---------------- batchnorm stats -> scale/shift ----------------
__global__ void bn_stats_kernel(const float* __restrict__ data, int C, int L,
                                const float* __restrict__ g, const float* __restrict__ beta,
                                float* scale, float* shift) {
  int c = blockIdx.x;
  __shared__ float ssum[256], ssq[256];
  float s = 0.f, q = 0.f;
  int total = N_NODES * L;
  for (int i = threadIdx.x; i < total; i += blockDim.x) {
    int n = i / L, l = i % L;
    float v = data[(n * C + c) * L + l];
    s += v; q += v * v;
  }
  ssum[threadIdx.x] = s; ssq[threadIdx.x] = q;
  __syncthreads();
  for (int o = 128; o > 0; o >>= 1) {
    if (threadIdx.x < o) { ssum[threadIdx.x] += ssum[threadIdx.x + o]; ssq[threadIdx.x] += ssq[threadIdx.x + o]; }
    __syncthreads();
  }
  if (threadIdx.x == 0) {
    float m = ssum[0] / (float)total;
    float var = ssq[0] / (float)total - m * m;
    float sc = g[c] * rsqrtf(var + EPSF);
    scale[c] = sc;
    shift[c] = beta[c] - m * sc;
  }
}

// ---------------- conv2 (+bn1 fused on input) + relu ----------------
__global__ void conv2_kernel(const float* __restrict__ A, const float* __restrict__ scale1,
                             const float* __restrict__ shift1, const float* __restrict__ w2,
                             const float* __restrict__ b2, float* __restrict__ B) {
  int gid = blockIdx.x * blockDim.x + threadIdx.x;   // 512*32
  int n = gid >> 5, r = gid & 31;
  int c2 = r >> 1, p = r & 1;
  float acc = b2[c2];
  for (int c1 = 0; c1 < C1C; c1++) {
    float sc = scale1[c1], sh = shift1[c1];
#pragma unroll
    for (int k = 0; k < K2C; k++) {
      float v = A[(n * C1C + c1) * L1C + p * S2C + k] * sc + sh;
      acc += w2[(c2 * C1C + c1) * K2C + k] * v;
    }
  }
  B[(n * C2C + c2) * L2C + p] = fmaxf(acc, 0.f);
}

// ---------------- bn2 apply -> x (512,32) ----------------
__global__ void xbuild_kernel(const float* __restrict__ B, const float* __restrict__ scale2,
                              const float* __restrict__ shift2, float* __restrict__ x) {
  int i = blockIdx.x * blockDim.x + threadIdx.x;   // 16384
  int e = i & 31;
  int c = e >> 1;
  x[i] = B[i] * scale2[c] + shift2[c];
}

// ---------------- diffusion state prep ----------------
// S row layout per node (KSTACK=320): [xh(64) | t1o(64) | t1i(64) | t2o(64) | t2i(64)]
__global__ void prep1_kernel(const float* __restrict__ x, const float* __restrict__ H, float* S) {
  int i = blockIdx.x * blockDim.x + threadIdx.x;   // 512*64
  int n = i >> 6, f = i & 63;
  float v = (f < 32) ? x[n * 32 + f] : H[n * 32 + (f - 32)];
  float* row = S + n * KSTACK;
  row[f] = v; row[64 + f] = 0.f; row[128 + f] = 0.f; row[192 + f] = -v; row[256 + f] = -v;
}
__global__ void prep2_kernel(float* S) {
  int i = blockIdx.x * blockDim.x + threadIdx.x;   // 512*64
  int n = i >> 6, f = i & 63;
  float* row = S + n * KSTACK;
  float v = row[f];
  row[64 + f] = 0.f; row[128 + f] = 0.f; row[192 + f] = -v; row[256 + f] = -v;
}

// ---------------- graph scatter (segment sums) ----------------
__global__ void scatter1_kernel(const int* __restrict__ ei, const float* __restrict__ dinv_out,
                                const float* __restrict__ dinv_in, float* S) {
  int e = blockIdx.x, f = threadIdx.x;             // 26000 x 64
  int s = ei[e], d = ei[N_EDGES + e];
  atomicAdd(&S[d * KSTACK + 64 + f], S[s * KSTACK + f] * dinv_out[s]);   // t1o += P_fwd(xh)
  atomicAdd(&S[s * KSTACK + 128 + f], S[d * KSTACK + f] * dinv_in[d]);   // t1i += P_bwd(xh)
}
__global__ void scatter2_kernel(const int* __restrict__ ei, const float* __restrict__ dinv_out,
                                const float* __restrict__ dinv_in, float* S) {
  int e = blockIdx.x, f = threadIdx.x;
  int s = ei[e], d = ei[N_EDGES + e];
  atomicAdd(&S[d * KSTACK + 192 + f], 2.f * S[s * KSTACK + 64 + f] * dinv_out[s]);  // t2o = 2*P_fwd(t1o) - xh
  atomicAdd(&S[s * KSTACK + 256 + f], 2.f * S[d * KSTACK + 128 + f] * dinv_in[d]);  // t2i = 2*P_bwd(t1i) - xh
}

// ---------------- WMMA f32 GEMM: (512 x 320) @ (320 x Ncols) ----------------
// One wave32 per 16x16 output tile; K-loop of V_WMMA_F32_16X16X4_F32.
__global__ void wmma_matmul_kernel(const float* __restrict__ Smat, const float* __restrict__ W,
                                   float* __restrict__ Out, int Ncols) {
  int ntiles = Ncols >> 4;
  int tileN = blockIdx.x % ntiles;
  int tileM = blockIdx.x / ntiles;
  int lane = threadIdx.x;
  int half = lane >> 4;      // 0: lanes 0-15, 1: lanes 16-31
  int lmod = lane & 15;
  int m = tileM * 16 + lmod;
  int n = tileN * 16 + lmod;
  v8f acc = {};
#if __has_builtin(__builtin_amdgcn_wmma_f32_16x16x4_f32)
  const float* arow = Smat + m * KSTACK + half * 2;   // A 16x4: lane<16 -> K=k,k+1 ; lane>=16 -> K=k+2,k+3
  for (int k = 0; k < KSTACK; k += 4) {
    v2f a; a.x = arow[k]; a.y = arow[k + 1];
    int kb = k + half * 2;                            // B 4x16 row-striped: VGPR0 K=kb, VGPR1 K=kb+1
    v2f bf; bf.x = W[kb * Ncols + n]; bf.y = W[(kb + 1) * Ncols + n];
    acc = __builtin_amdgcn_wmma_f32_16x16x4_f32(false, a, false, bf, (short)0, acc,
                                                false, false);
  }
#else
  // scalar fallback with identical output ownership
  for (int v = 0; v < 8; v++) {
    int mr = tileM * 16 + v + 8 * half;
    float s = 0.f;
    for (int k = 0; k < KSTACK; k++) s += Smat[mr * KSTACK + k] * W[k * Ncols + n];
    acc[v] = s;
  }
#endif
  // D layout: VGPR v, lanes 0-15 -> M = v ; lanes 16-31 -> M = v + 8
#pragma unroll
  for (int v = 0; v < 8; v++) Out[(tileM * 16 + v + 8 * half) * Ncols + n] = acc[v];
}

// ---------------- GRU gates: Z,R + build xh2 into S[:, :64] ----------------
__global__ void gate_kernel(const float* __restrict__ G, const float* __restrict__ bz,
                            const float* __restrict__ br, const float* __restrict__ x,
                            const float* __restrict__ H, float* __restrict__ Z, float* S) {
  int i = blockIdx.x * blockDim.x + threadIdx.x;   // 16384
  int n = i >> 5, c = i & 31;
  float zp = G[n * 64 + c] + bz[c];
  float rp = G[n * 64 + 32 + c] + br[c];
  float z = 1.f / (1.f + expf(-zp));
  float r = 1.f / (1.f + expf(-rp));
  Z[i] = z;
  S[n * KSTACK + c] = x[i];
  S[n * KSTACK + 32 + c] = r * H[i];
}

// ---------------- GRU update: H = relu(Z*H + (1-Z)*tanh(Hpre+bh)) ----------------
__global__ void update_kernel(const float* __restrict__ Hpre, const float* __restrict__ bh,
                              const float* __restrict__ Z, float* H) {
  int i = blockIdx.x * blockDim.x + threadIdx.x;   // 16384
  int c = i & 31;
  float ht = tanhf(Hpre[i] + bh[c]);
  float z = Z[i];
  float h = z * H[i] + (1.f - z) * ht;
  H[i] = fmaxf(h, 0.f);
}

// ---------------- streaming matvec: out[r] = W[r,:] . v + bias[r] ----------------
// One wave32 per row, float4 (b128) loads; HBM-bound head (lin1/cls/reg).
__global__ void matvec_kernel(const float* __restrict__ W, const float* __restrict__ v,
                              const float* __restrict__ bias, float* __restrict__ out,
                              int rows, int cols) {
  int wave = threadIdx.x >> 5;
  int lane = threadIdx.x & 31;
  int r = blockIdx.x * (blockDim.x >> 5) + wave;
  if (r >= rows) return;
  const float* wr = W + (size_t)r * (size_t)cols;
  float acc = 0.f;
  for (int j = lane * 4; j < cols; j += 128) {
    float4 a = *(const float4*)(wr + j);
    float4 b = *(const float4*)(v + j);
    acc += a.x * b.x + a.y * b.y + a.z * b.z + a.w * b.w;
  }
#pragma unroll
  for (int o = 16; o > 0; o >>= 1) acc += __shfl_down(acc, o, 32);
  if (lane == 0) out[r] = acc + bias[r];
}

extern "C" void kernel_launch(void* const* d_in, const int* in_sizes, int n_in,
                              void* d_out, int out_size, void* d_ws, size_t ws_size,
                              hipStream_t stream) {
  (void)in_sizes; (void)n_in; (void)out_size; (void)ws_size;
  const float* inputs  = (const float*)d_in[0];
  const int*   ei      = (const int*)d_in[1];
  const float* conv1_w = (const float*)d_in[2];
  const float* conv1_b = (const float*)d_in[3];
  const float* bn1_g   = (const float*)d_in[4];
  const float* bn1_b   = (const float*)d_in[5];
  const float* conv2_w = (const float*)d_in[6];
  const float* conv2_b = (const float*)d_in[7];
  const float* bn2_g   = (const float*)d_in[8];
  const float* bn2_b   = (const float*)d_in[9];
  const float* wz      = (const float*)d_in[10];
  const float* bz      = (const float*)d_in[11];
  const float* wr      = (const float*)d_in[12];
  const float* br      = (const float*)d_in[13];
  const float* wh      = (const float*)d_in[14];
  const float* bh      = (const float*)d_in[15];
  const float* lin1_w  = (const float*)d_in[16];
  const float* lin1_b  = (const float*)d_in[17];
  const float* reg_w   = (const float*)d_in[18];
  const float* reg_b   = (const float*)d_in[19];
  const float* cls1_w  = (const float*)d_in[20];
  const float* cls1_b  = (const float*)d_in[21];
  const float* cls2_w  = (const float*)d_in[22];
  const float* cls2_b  = (const float*)d_in[23];
  float* dout = (float*)d_out;

  // ---- workspace carve-up (all offsets multiples of 16 floats -> 64B aligned) ----
  float* ws = (float*)d_ws;
  size_t o = 0;
  float* dinv_out = ws + o; o += 512;          // contiguous with dinv_in for joint zero/inv
  float* dinv_in  = ws + o; o += 512;
  float* scale1   = ws + o; o += 32;
  float* shift1   = ws + o; o += 32;
  float* scale2   = ws + o; o += 16;
  float* shift2   = ws + o; o += 16;
  float* Wzr      = ws + o; o += KSTACK * 64;  // 320x64 (Z|R)
  float* Whc      = ws + o; o += KSTACK * 32;  // 320x32
  float* A        = ws + o; o += N_NODES * C1C * L1C;   // 245760
  float* Bbuf     = ws + o; o += N_NODES * 32;
  float* xbuf     = ws + o; o += N_NODES * EMB;
  float* H        = ws + o; o += N_NODES * EMB;
  float* Zbuf     = ws + o; o += N_NODES * EMB;
  float* S        = ws + o; o += N_NODES * KSTACK;      // 163840
  float* G        = ws + o; o += N_NODES * 64;
  float* Hpre     = ws + o; o += N_NODES * EMB;
  float* outvec   = ws + o; o += H2DIM;
  float* tmpc     = ws + o; o += H8DIM;

  // ---- graph degrees -> dinv ----
  zero_kernel<<<4, 256, 0, stream>>>(dinv_out, 1024);          // covers dinv_out+dinv_in
  deg_kernel<<<(N_EDGES + 255) / 256, 256, 0, stream>>>(ei, dinv_out, dinv_in);
  inv_kernel<<<4, 256, 0, stream>>>(dinv_out, 1024);
  // ---- combined diffusion weights ----
  wcomb_kernel<<<(KSTACK * 96 + 255) / 256, 256, 0, stream>>>(wz, wr, wh, Wzr, Whc);
  // ---- H0 = 0 ----
  zero_kernel<<<64, 256, 0, stream>>>(H, N_NODES * EMB);

  // ---- 63 recurrent windows ----
  for (int t = 0; t < N_WIN; t++) {
    int start = t * SLIDE;
    conv1_kernel<<<N_NODES, C1C * L1C, 0, stream>>>(inputs, conv1_w, conv1_b, A, start);
    bn_stats_kernel<<<C1C, 256, 0, stream>>>(A, C1C, L1C, bn1_g, bn1_b, scale1, shift1);
    conv2_kernel<<<64, 256, 0, stream>>>(A, scale1, shift1, conv2_w, conv2_b, Bbuf);
    bn_stats_kernel<<<C2C, 256, 0, stream>>>(Bbuf, C2C, L2C, bn2_g, bn2_b, scale2, shift2);
    xbuild_kernel<<<64, 256, 0, stream>>>(Bbuf, scale2, shift2, xbuf);

    // Z/R path: xh = [x, H]
    prep1_kernel<<<128, 256, 0, stream>>>(xbuf, H, S);
    scatter1_kernel<<<N_EDGES, 64, 0, stream>>>(ei, dinv_out, dinv_in, S);
    scatter2_kernel<<<N_EDGES, 64, 0, stream>>>(ei, dinv_out, dinv_in, S);
    wmma_matmul_kernel<<<32 * 4, 32, 0, stream>>>(S, Wzr, G, 64);
    gate_kernel<<<64, 256, 0, stream>>>(G, bz, br, xbuf, H, Zbuf, S);

    // H~ path: xh2 = [x, R*H] (already in S[:, :64])
    prep2_kernel<<<128, 256, 0, stream>>>(S);
    scatter1_kernel<<<N_EDGES, 64, 0, stream>>>(ei, dinv_out, dinv_in, S);
    scatter2_kernel<<<N_EDGES, 64, 0, stream>>>(ei, dinv_out, dinv_in, S);
    wmma_matmul_kernel<<<32 * 2, 32, 0, stream>>>(S, Whc, Hpre, 32);
    update_kernel<<<64, 256, 0, stream>>>(Hpre, bh, Zbuf, H);
  }

  // ---- readout head (HBM-bound streaming matvecs) ----
  matvec_kernel<<<H2DIM / 8, 256, 0, stream>>>(lin1_w, H, lin1_b, outvec, H2DIM, N_NODES * EMB);
  matvec_kernel<<<(REGROWS + 7) / 8, 256, 0, stream>>>(reg_w, outvec, reg_b, dout, REGROWS, H2DIM);
  matvec_kernel<<<H8DIM / 8, 256, 0, stream>>>(cls1_w, outvec, cls1_b, tmpc, H8DIM, H2DIM);
  matvec_kernel<<<1, 256, 0, stream>>>(cls2_w, tmpc, cls2_b, dout + REGROWS, CLS2ROWS, H8DIM);
}